// SelfAttention_88845693485061
// MI455X (gfx1250) — compile-verified
//
#include <hip/hip_runtime.h>
#include <hip/hip_bf16.h>

// Problem constants (match the reference)
#define Bq 4
#define Nq 2048
#define Dq 512
#define DHq 128
#define Hq 6
#define HDq (Hq * DHq)   // 768
#define ROWS (Bq * Nq)   // 8192
#define KSTRIDE 132      // padded LDS row stride (floats): lane m -> bank 4m, conflict-free

typedef __attribute__((ext_vector_type(2))) float v2f;
typedef __attribute__((ext_vector_type(4))) float v4f;
typedef __attribute__((ext_vector_type(8))) float v8f;
typedef __attribute__((ext_vector_type(4))) int   v4i;

#if __has_builtin(__builtin_amdgcn_global_load_async_to_lds_b128) && \
    __has_builtin(__builtin_amdgcn_s_wait_asynccnt)
#define USE_ASYNC_LDS 1
// Builtin prototype (from clang diagnostic): (v4i addrspace(1)*, v4i addrspace(3)*, imm, imm)
#define GPTR(p) ((__attribute__((address_space(1))) v4i*)(p))
#define LPTR(p) ((__attribute__((address_space(3))) v4i*)(p))
#endif

// D = A(16x4 f32) * B(4x16 f32) + C(16x16 f32), full fp32 precision on the matrix pipe.
__device__ __forceinline__ v8f wmma_f32(v2f a, v2f b, v8f c) {
    return __builtin_amdgcn_wmma_f32_16x16x4_f32(
        /*neg_a=*/false, a, /*neg_b=*/false, b,
        /*c_mod=*/(short)0, c, /*reuse_a=*/false, /*reuse_b=*/false);
}

// ---------------------------------------------------------------------------
// Kernel 1: Y(ROWS x HD) = X(ROWS x D) @ W(D x HD) + bias   (q, k, v)
// Each wave computes a 16x64 strip: one A fragment feeds 4 WMMAs per k-step.
// grid = (HD/64, ROWS/(16*8)), block = 256
// ---------------------------------------------------------------------------
__global__ void __launch_bounds__(256)
proj_gemm_kernel(const float* __restrict__ X, const float* __restrict__ W,
                 const float* __restrict__ bias, float* __restrict__ Y) {
    const int lane = threadIdx.x & 31;
    const int wave = threadIdx.x >> 5;
    const int row0 = (blockIdx.y * 8 + wave) * 16;
    const int col0 = blockIdx.x * 64;
    const int m    = lane & 15;
    const int half = lane >> 4;

    const float* __restrict__ arow = X + (size_t)(row0 + m) * Dq;
    const float* __restrict__ bbase = W + col0 + m;

    v8f acc[4] = {{}, {}, {}, {}};
    #pragma unroll 2
    for (int k0 = 0; k0 < Dq; k0 += 4) {
        const int ka = k0 + half * 2;
        const v2f a = *(const v2f*)(arow + ka);
        const float* __restrict__ bk0 = bbase + (size_t)ka * HDq;
        const float* __restrict__ bk1 = bbase + (size_t)(ka + 1) * HDq;
        #pragma unroll
        for (int j = 0; j < 4; ++j) {
            v2f b;
            b.x = bk0[j * 16];
            b.y = bk1[j * 16];
            acc[j] = wmma_f32(a, b, acc[j]);
        }
    }

    #pragma unroll
    for (int j = 0; j < 4; ++j) {
        const float bvj = bias[col0 + j * 16 + m];
        float* __restrict__ yb = Y + (size_t)row0 * HDq + col0 + j * 16 + m;
        #pragma unroll
        for (int r = 0; r < 8; ++r) {
            const int mm = r + half * 8;
            yb[(size_t)mm * HDq] = acc[j][r] + bvj;
        }
    }
}

// ---------------------------------------------------------------------------
// Kernel 2: per (b,h,t): streaming logsumexp over u of q.k/sqrt(D) + diagonal
// score -> diag_prob[b,t,h]. K tiles DMA'd into double-buffered LDS with
// GLOBAL_LOAD_ASYNC_TO_LDS_B128 (ASYNCcnt-tracked), per-lane online stats
// merged by one butterfly at the end, Q fragment resident in registers.
// grid = (N/(16*8), H, B), block = 256
// ---------------------------------------------------------------------------
__global__ void __launch_bounds__(256)
attn_diag_kernel(const float* __restrict__ Q, const float* __restrict__ K,
                 float* __restrict__ diag_prob) {
#ifdef USE_ASYNC_LDS
    __shared__ float ksh[2][16 * KSTRIDE];
#else
    __shared__ float ksh1[16 * KSTRIDE];
#endif

    const int tid   = threadIdx.x;
    const int lane  = tid & 31;
    const int wave  = tid >> 5;
    const int h     = blockIdx.y;
    const int b     = blockIdx.z;
    const int tTile = blockIdx.x * 8 + wave;
    const int t0    = tTile * 16;
    const int m     = lane & 15;
    const int half  = lane >> 4;
    const float scale = 0.04419417382415922f;             // 1/sqrt(512)

    const size_t hoff = (size_t)h * DHq;
    const float* __restrict__ qrow  = Q + ((size_t)b * Nq + t0 + m) * HDq + hoff;
    const float* __restrict__ kbase = K + ((size_t)b * Nq) * HDq + hoff;

    // Hoist the whole 16x128 Q fragment into registers (64 VGPRs/lane).
    v2f qa[DHq / 4];
    #pragma unroll
    for (int i = 0; i < DHq / 4; ++i)
        qa[i] = *(const v2f*)(qrow + i * 4 + half * 2);

    // Per-lane online-softmax partials: lane's column slot of row M=r+8*half.
    float m_lane[8], l_lane[8], diag_s[8];
    #pragma unroll
    for (int r = 0; r < 8; ++r) { m_lane[r] = -3.0e38f; l_lane[r] = 0.0f; diag_s[r] = 0.0f; }

    // Cooperative K-tile copy pattern: thread -> (row, 8-float chunk).
    const int crow = tid >> 4;
    const int ccol = (tid & 15) * 8;
    const float* __restrict__ gk = kbase + (size_t)crow * HDq + ccol;
    const int NT = Nq / 16;

#ifdef USE_ASYNC_LDS
    // ---- async global->LDS double-buffered pipeline ----
    {
        float* l0 = &ksh[0][crow * KSTRIDE + ccol];
        __builtin_amdgcn_global_load_async_to_lds_b128(GPTR(gk),     LPTR(l0),     0, 0);
        __builtin_amdgcn_global_load_async_to_lds_b128(GPTR(gk + 4), LPTR(l0 + 4), 0, 0);
    }
    for (int u = 0; u < NT; ++u) {
        if (u + 1 < NT) {
            const float* g = gk + (size_t)(u + 1) * 16 * HDq;
            float* l = &ksh[(u + 1) & 1][crow * KSTRIDE + ccol];
            __builtin_amdgcn_global_load_async_to_lds_b128(GPTR(g),     LPTR(l),     0, 0);
            __builtin_amdgcn_global_load_async_to_lds_b128(GPTR(g + 4), LPTR(l + 4), 0, 0);
            if (u + 2 < NT)
                __builtin_prefetch(gk + (size_t)(u + 2) * 16 * HDq, 0, 0);
            __builtin_amdgcn_s_wait_asynccnt(2);          // tile u landed in LDS
        } else {
            __builtin_amdgcn_s_wait_asynccnt(0);
        }
        __syncthreads();

        const float* __restrict__ kb = &ksh[u & 1][0];
        v8f acc = {};
        #pragma unroll 8
        for (int k0 = 0; k0 < DHq; k0 += 4) {
            const int ka = k0 + half * 2;
            const v2f bb = *(const v2f*)&kb[m * KSTRIDE + ka];   // K[u0+n=m][ka..ka+1]
            acc = wmma_f32(qa[k0 >> 2], bb, acc);
        }

        #pragma unroll
        for (int r = 0; r < 8; ++r) {
            const float s = acc[r] * scale;
            const float newm = fmaxf(m_lane[r], s);
            l_lane[r] = l_lane[r] * __expf(m_lane[r] - newm) + __expf(s - newm);
            m_lane[r] = newm;
            if (u == tTile && m == r + half * 8)          // diagonal element s[M][M]
                diag_s[r] = s;
        }
        __syncthreads();
    }
#else
    // ---- fallback: register-staged single-buffer pipeline ----
    v4f r0 = *(const v4f*)(gk);
    v4f r1 = *(const v4f*)(gk + 4);
    for (int u = 0; u < NT; ++u) {
        *(v4f*)&ksh1[crow * KSTRIDE + ccol]     = r0;
        *(v4f*)&ksh1[crow * KSTRIDE + ccol + 4] = r1;
        __syncthreads();

        if (u + 1 < NT) {
            const float* __restrict__ g = gk + (size_t)(u + 1) * 16 * HDq;
            r0 = *(const v4f*)(g);
            r1 = *(const v4f*)(g + 4);
            if (u + 2 < NT)
                __builtin_prefetch(gk + (size_t)(u + 2) * 16 * HDq, 0, 0);
        }

        v8f acc = {};
        #pragma unroll 8
        for (int k0 = 0; k0 < DHq; k0 += 4) {
            const int ka = k0 + half * 2;
            const v2f bb = *(const v2f*)&ksh1[m * KSTRIDE + ka];
            acc = wmma_f32(qa[k0 >> 2], bb, acc);
        }

        #pragma unroll
        for (int r = 0; r < 8; ++r) {
            const float s = acc[r] * scale;
            const float newm = fmaxf(m_lane[r], s);
            l_lane[r] = l_lane[r] * __expf(m_lane[r] - newm) + __expf(s - newm);
            m_lane[r] = newm;
            if (u == tTile && m == r + half * 8)
                diag_s[r] = s;
        }
        __syncthreads();
    }
#endif

    // One butterfly merge of (max, sumexp) partials across the 16-lane half.
    #pragma unroll
    for (int r = 0; r < 8; ++r) {
        float mm = m_lane[r], ll = l_lane[r];
        #pragma unroll
        for (int mask = 1; mask <= 8; mask <<= 1) {
            const float om = __shfl_xor(mm, mask, 32);
            const float ol = __shfl_xor(ll, mask, 32);
            const float nm = fmaxf(mm, om);
            ll = ll * __expf(mm - nm) + ol * __expf(om - nm);
            mm = nm;
        }
        const int M = r + half * 8;
        if (m == M) {                                      // lane holding row M's diagonal
            const int t = t0 + M;
            diag_prob[((size_t)b * Nq + t) * Hq + h] = __expf(diag_s[r] - mm) / ll;
        }
    }
}

// ---------------------------------------------------------------------------
// Kernel 3: out(ROWS x D) = (diag_prob ⊙ V)(ROWS x HD) @ Wo(HD x D) + bo
// 16x64 strip per wave; diag_prob scaling fused into the A-matrix load.
// grid = (D/64, ROWS/(16*8)), block = 256
// ---------------------------------------------------------------------------
__global__ void __launch_bounds__(256)
out_gemm_kernel(const float* __restrict__ V, const float* __restrict__ dp,
                const float* __restrict__ Wo, const float* __restrict__ bo,
                float* __restrict__ out) {
    const int lane = threadIdx.x & 31;
    const int wave = threadIdx.x >> 5;
    const int row0 = (blockIdx.y * 8 + wave) * 16;
    const int col0 = blockIdx.x * 64;
    const int m    = lane & 15;
    const int half = lane >> 4;

    const float* __restrict__ arow  = V  + (size_t)(row0 + m) * HDq;
    const float* __restrict__ dprow = dp + (size_t)(row0 + m) * Hq;
    const float* __restrict__ bbase = Wo + col0 + m;

    v8f acc[4] = {{}, {}, {}, {}};
    #pragma unroll 2
    for (int k0 = 0; k0 < HDq; k0 += 4) {
        const int ka = k0 + half * 2;
        v2f a = *(const v2f*)(arow + ka);
        const float s = dprow[ka >> 7];                   // head = ka / DH, same for ka..ka+1
        a.x *= s; a.y *= s;
        const float* __restrict__ bk0 = bbase + (size_t)ka * Dq;
        const float* __restrict__ bk1 = bbase + (size_t)(ka + 1) * Dq;
        #pragma unroll
        for (int j = 0; j < 4; ++j) {
            v2f b;
            b.x = bk0[j * 16];
            b.y = bk1[j * 16];
            acc[j] = wmma_f32(a, b, acc[j]);
        }
    }

    #pragma unroll
    for (int j = 0; j < 4; ++j) {
        const float bvj = bo[col0 + j * 16 + m];
        float* __restrict__ ob = out + (size_t)row0 * Dq + col0 + j * 16 + m;
        #pragma unroll
        for (int r = 0; r < 8; ++r) {
            const int mm = r + half * 8;
            ob[(size_t)mm * Dq] = acc[j][r] + bvj;
        }
    }
}

extern "C" void kernel_launch(void* const* d_in, const int* in_sizes, int n_in,
                              void* d_out, int out_size, void* d_ws, size_t ws_size,
                              hipStream_t stream) {
    (void)in_sizes; (void)n_in; (void)out_size; (void)ws_size;
    const float* emb = (const float*)d_in[0];
    const float* Wq  = (const float*)d_in[1];
    const float* bq  = (const float*)d_in[2];
    const float* Wk  = (const float*)d_in[3];
    const float* bk  = (const float*)d_in[4];
    const float* Wv  = (const float*)d_in[5];
    const float* bv  = (const float*)d_in[6];
    const float* Wo  = (const float*)d_in[7];
    const float* bo  = (const float*)d_in[8];
    float* outp = (float*)d_out;

    // Workspace: q, k, v (ROWS x HD each) + diag_prob (ROWS x H)
    float* q  = (float*)d_ws;
    float* k  = q + (size_t)ROWS * HDq;
    float* v  = k + (size_t)ROWS * HDq;
    float* dp = v + (size_t)ROWS * HDq;

    const dim3 blk(256);
    const dim3 gProj(HDq / 64, ROWS / (16 * 8));          // (12, 64)
    proj_gemm_kernel<<<gProj, blk, 0, stream>>>(emb, Wq, bq, q);
    proj_gemm_kernel<<<gProj, blk, 0, stream>>>(emb, Wk, bk, k);
    proj_gemm_kernel<<<gProj, blk, 0, stream>>>(emb, Wv, bv, v);

    const dim3 gAttn(Nq / (16 * 8), Hq, Bq);              // (16, 6, 4)
    attn_diag_kernel<<<gAttn, blk, 0, stream>>>(q, k, dp);

    const dim3 gOut(Dq / 64, ROWS / (16 * 8));            // (8, 64)
    out_gemm_kernel<<<gOut, blk, 0, stream>>>(v, dp, Wo, bo, outp);
}